// MultivariateAttention_6871947673879
// MI455X (gfx1250) — compile-verified
//
#include <hip/hip_runtime.h>

// ---------------------------------------------------------------------------
// MultivariateAttention for MI455X (gfx1250), fp32 end-to-end using
// V_WMMA_F32_16X16X4_F32 for every GEMM stage.
// ---------------------------------------------------------------------------

typedef float v2f __attribute__((ext_vector_type(2)));
typedef float v8f __attribute__((ext_vector_type(8)));

constexpr int kB = 4, kS = 4, kC = 4, kL = 512, kE = 64, kA = 32, kH = 8;
constexpr int KTOT = kS * kL + kC;   // 2052 keys (sequences + constants)
constexpr int KPAD = 2064;           // padded to multiple of 16
constexpr int kF   = kH * kA;        // 256 head features

// Workspace layout (floats)
constexpr size_t WSQ_N  = (size_t)kB * kS * kH * kL * kA;   // q_s   [B,S,H,L,A]
constexpr size_t WSK_N  = (size_t)kB * kH * KPAD * kA;      // k_all [B,H,KPAD,A]
constexpr size_t WSV_N  = (size_t)kB * kH * KPAD * kA;      // v_all [B,H,KPAD,A]
constexpr size_t WSO_N  = (size_t)kB * kS * kL * kF;        // out_s [B,S,L,H*A]
constexpr size_t WSQC_N = (size_t)kB * kC * kH * kA;        // q_c   [B,C,H,A]
constexpr size_t WSOC_N = (size_t)kB * kC * kH * kA;        // out_c [B,C,H*A]

__device__ __forceinline__ v8f wmma4(v2f a, v2f b, v8f c) {
  // (neg_a, A, neg_b, B, c_mod, C, reuse_a, reuse_b)
  return __builtin_amdgcn_wmma_f32_16x16x4_f32(false, a, false, b, (short)0, c,
                                               false, false);
}

// ---------------------------------------------------------------------------
// Kernel 1: sequence QKV projections.  Y = X[b,s] (512x64) @ W[s,h] (64x32)
// grid.x = 3*B*S*H, 256 threads (8 waves), each wave owns L/16/8 = 4 M-tiles.
// ---------------------------------------------------------------------------
__global__ void proj_seq_kernel(const float* __restrict__ Xq,
                                const float* __restrict__ Xk,
                                const float* __restrict__ Xv,
                                const float* __restrict__ Wq,
                                const float* __restrict__ Wk,
                                const float* __restrict__ Wv,
                                float* __restrict__ wsq,
                                float* __restrict__ wsk,
                                float* __restrict__ wsv) {
  int id = blockIdx.x;
  int p  = id / (kB * kS * kH);
  int rm = id % (kB * kS * kH);
  int bb = rm / (kS * kH);
  int sh = rm % (kS * kH);
  int s  = sh / kH, h = sh % kH;

  const float* X = (p == 0 ? Xq : (p == 1 ? Xk : Xv)) + (size_t)(bb * kS + s) * kL * kE;
  const float* W = (p == 0 ? Wq : (p == 1 ? Wk : Wv)) + (size_t)(s * kH + h) * kE * kA;

  int wave = threadIdx.x >> 5, lane = threadIdx.x & 31;
  int half = lane >> 4, lid = lane & 15;

  for (int mt = wave; mt < kL / 16; mt += 8) {
    int m0 = mt * 16;
    v8f acc0 = {}, acc1 = {};
    for (int k0 = 0; k0 < kE; k0 += 4) {
      int kk = k0 + 2 * half;
      v2f a;  a.x = X[(size_t)(m0 + lid) * kE + kk];     a.y = X[(size_t)(m0 + lid) * kE + kk + 1];
      v2f b0; b0.x = W[(size_t)kk * kA + lid];           b0.y = W[(size_t)(kk + 1) * kA + lid];
      v2f b1; b1.x = W[(size_t)kk * kA + 16 + lid];      b1.y = W[(size_t)(kk + 1) * kA + 16 + lid];
      acc0 = wmma4(a, b0, acc0);
      acc1 = wmma4(a, b1, acc1);
    }
    for (int r = 0; r < 8; ++r) {
      int m = r + 8 * half, l = m0 + m;
      size_t off;
      float* dst;
      if (p == 0) { dst = wsq; off = (((size_t)(bb * kS + s) * kH + h) * kL + l) * kA; }
      else        { dst = (p == 1 ? wsk : wsv);
                    off = ((size_t)(bb * kH + h) * KPAD + (size_t)s * kL + l) * kA; }
      dst[off + lid]      = acc0[r];
      dst[off + 16 + lid] = acc1[r];
    }
  }
}

// ---------------------------------------------------------------------------
// Kernel 2: constant projections (1x64x32 each) + zero the KPAD tail rows.
// grid.x = B*C*H, 96 threads.
// ---------------------------------------------------------------------------
__global__ void proj_con_kernel(const float* __restrict__ Xq,
                                const float* __restrict__ Xk,
                                const float* __restrict__ Xv,
                                const float* __restrict__ Wq,
                                const float* __restrict__ Wk,
                                const float* __restrict__ Wv,
                                float* __restrict__ wsqc,
                                float* __restrict__ wsk,
                                float* __restrict__ wsv) {
  int id = blockIdx.x;
  int bb = id / (kC * kH);
  int ch = id % (kC * kH);
  int c  = ch / kH, h = ch % kH;
  int t  = threadIdx.x;
  int p  = t / kA, a = t % kA;

  if (t < 96) {
    const float* X = (p == 0 ? Xq : (p == 1 ? Xk : Xv)) + (size_t)(bb * kC + c) * kE;
    const float* W = (p == 0 ? Wq : (p == 1 ? Wk : Wv)) + (size_t)(c * kH + h) * kE * kA;
    float acc = 0.f;
    for (int e = 0; e < kE; ++e) acc += X[e] * W[(size_t)e * kA + a];
    if (p == 0) {
      wsqc[((size_t)(bb * kC + c) * kH + h) * kA + a] = acc;
    } else {
      float* dst = (p == 1 ? wsk : wsv);
      dst[((size_t)(bb * kH + h) * KPAD + (size_t)kS * kL + c) * kA + a] = acc;
    }
  }
  if (c == 0) {  // zero pad rows KTOT..KPAD-1 once per (b,h)
    for (int i = t; i < (KPAD - KTOT) * kA; i += blockDim.x) {
      int row = i / kA, col = i % kA;
      size_t off = ((size_t)(bb * kH + h) * KPAD + KTOT + row) * kA + col;
      wsk[off] = 0.f;
      wsv[off] = 0.f;
    }
  }
}

// ---------------------------------------------------------------------------
// Kernel 3: flash attention per (b,s,h).  grid.x = B*S*H, 256 threads.
// Each wave owns 16-row query tiles; online softmax over 129 key tiles of 16.
// ---------------------------------------------------------------------------
__global__ void attn_seq_kernel(const float* __restrict__ wsq,
                                const float* __restrict__ wsk,
                                const float* __restrict__ wsv,
                                float* __restrict__ wso,
                                const int* __restrict__ maskp) {
  int id = blockIdx.x;
  int bb = id / (kS * kH);
  int sh = id % (kS * kH);
  int s  = sh / kH, h = sh % kH;
  bool use_mask = (maskp[0] != 0);

  const float* Q  = wsq + ((size_t)(bb * kS + s) * kH + h) * kL * kA;
  const float* Km = wsk + (size_t)(bb * kH + h) * KPAD * kA;
  const float* Vm = wsv + (size_t)(bb * kH + h) * KPAD * kA;

  __shared__ float pbuf[8][16][17];  // per-wave P-tile transpose buffer

  int wave = threadIdx.x >> 5, lane = threadIdx.x & 31;
  int half = lane >> 4, lid = lane & 15;
  const float scale = 0.17677669529663687f;  // 1/sqrt(32)

  for (int mt = wave; mt < kL / 16; mt += 8) {
    int m0 = mt * 16;

    v2f qa[8];  // A-fragments of the 16x32 Q tile (K chunks of 4)
    for (int cc = 0; cc < 8; ++cc) {
      int kk = cc * 4 + 2 * half;
      qa[cc].x = Q[(size_t)(m0 + lid) * kA + kk];
      qa[cc].y = Q[(size_t)(m0 + lid) * kA + kk + 1];
    }

    v8f o0 = {}, o1 = {};
    float rmax[8], rsum[8];
    for (int r = 0; r < 8; ++r) { rmax[r] = -1e30f; rsum[r] = 0.f; }

    for (int kt = 0; kt < KPAD / 16; ++kt) {
      int kt0 = kt * 16;
      int kcol = kt0 + lid;  // this lane's key column

      // scores tile = Q(16x32) @ K^T(32x16)
      v8f sc = {};
      for (int cc = 0; cc < 8; ++cc) {
        int kk = cc * 4 + 2 * half;
        v2f kb;
        kb.x = Km[(size_t)(kt0 + lid) * kA + kk];
        kb.y = Km[(size_t)(kt0 + lid) * kA + kk + 1];
        sc = wmma4(qa[cc], kb, sc);
      }

      // mask + online softmax update (per-row, 16-lane shuffle reductions)
      for (int r = 0; r < 8; ++r) {
        int m = r + 8 * half, lq = m0 + m;
        float sv = sc[r] * scale;
        bool dead = (kcol >= KTOT) ||
                    (use_mask && (kcol < kS * kL) && ((kcol & (kL - 1)) > lq));
        sv = dead ? -1e30f : sv;

        float tm = sv;
        tm = fmaxf(tm, __shfl_xor(tm, 1, 32));
        tm = fmaxf(tm, __shfl_xor(tm, 2, 32));
        tm = fmaxf(tm, __shfl_xor(tm, 4, 32));
        tm = fmaxf(tm, __shfl_xor(tm, 8, 32));
        float mnew = fmaxf(rmax[r], tm);

        float pv = dead ? 0.f : __expf(sv - mnew);
        float ps = pv;
        ps += __shfl_xor(ps, 1, 32);
        ps += __shfl_xor(ps, 2, 32);
        ps += __shfl_xor(ps, 4, 32);
        ps += __shfl_xor(ps, 8, 32);

        float corr = __expf(rmax[r] - mnew);
        rsum[r] = rsum[r] * corr + ps;
        rmax[r] = mnew;
        o0[r] *= corr;
        o1[r] *= corr;
        pbuf[wave][m][lid] = pv;  // D-layout -> LDS (row m, col lid)
      }
      __builtin_amdgcn_wave_barrier();  // DS ops in-order within wave

      // O += P(16x16) @ V(16x32)
      for (int cc = 0; cc < 4; ++cc) {
        int kk = cc * 4 + 2 * half;
        v2f pa;
        pa.x = pbuf[wave][lid][kk];
        pa.y = pbuf[wave][lid][kk + 1];
        v2f vb0, vb1;
        vb0.x = Vm[(size_t)(kt0 + kk) * kA + lid];
        vb0.y = Vm[(size_t)(kt0 + kk + 1) * kA + lid];
        vb1.x = Vm[(size_t)(kt0 + kk) * kA + 16 + lid];
        vb1.y = Vm[(size_t)(kt0 + kk + 1) * kA + 16 + lid];
        o0 = wmma4(pa, vb0, o0);
        o1 = wmma4(pa, vb1, o1);
      }
      __builtin_amdgcn_wave_barrier();  // reads done before next tile's writes
    }

    // normalize and write out_s [B,S,L,H*A]
    for (int r = 0; r < 8; ++r) {
      int m = r + 8 * half;
      float inv = 1.f / rsum[r];
      size_t off = ((size_t)(bb * kS + s) * kL + (m0 + m)) * kF + (size_t)h * kA;
      wso[off + lid]      = o0[r] * inv;
      wso[off + 16 + lid] = o1[r] * inv;
    }
  }
}

// ---------------------------------------------------------------------------
// Kernel 4: head mix (sequences): out[b,s] (512x64) = out_s (512x256) @ HW[s]
// grid.x = B*S, 256 threads.
// ---------------------------------------------------------------------------
__global__ void headmix_seq_kernel(const float* __restrict__ wso,
                                   const float* __restrict__ HW,
                                   float* __restrict__ out) {
  int id = blockIdx.x;
  int bb = id / kS, s = id % kS;
  const float* X = wso + (size_t)(bb * kS + s) * kL * kF;
  const float* W = HW + (size_t)s * kF * kE;

  int wave = threadIdx.x >> 5, lane = threadIdx.x & 31;
  int half = lane >> 4, lid = lane & 15;

  for (int mt = wave; mt < kL / 16; mt += 8) {
    int m0 = mt * 16;
    v8f acc[4] = {{}, {}, {}, {}};
    for (int cc = 0; cc < kF / 4; ++cc) {
      int kk = cc * 4 + 2 * half;
      v2f a;
      a.x = X[(size_t)(m0 + lid) * kF + kk];
      a.y = X[(size_t)(m0 + lid) * kF + kk + 1];
      for (int nt = 0; nt < 4; ++nt) {
        v2f b;
        b.x = W[(size_t)kk * kE + nt * 16 + lid];
        b.y = W[(size_t)(kk + 1) * kE + nt * 16 + lid];
        acc[nt] = wmma4(a, b, acc[nt]);
      }
    }
    for (int r = 0; r < 8; ++r) {
      int m = r + 8 * half;
      size_t off = ((size_t)(bb * kS + s) * kL + (m0 + m)) * kE;
      for (int nt = 0; nt < 4; ++nt) out[off + nt * 16 + lid] = acc[nt][r];
    }
  }
}

// ---------------------------------------------------------------------------
// Kernel 5: constants attention (16 query rows total) — VALU two-pass softmax.
// grid.x = B*C*H, 64 threads.
// ---------------------------------------------------------------------------
__global__ void attn_con_kernel(const float* __restrict__ wsqc,
                                const float* __restrict__ wsk,
                                const float* __restrict__ wsv,
                                float* __restrict__ wsoc) {
  int id = blockIdx.x;
  int bb = id / (kC * kH);
  int ch = id % (kC * kH);
  int c  = ch / kH, h = ch % kH;

  const float* q  = wsqc + ((size_t)(bb * kC + c) * kH + h) * kA;
  const float* Km = wsk + (size_t)(bb * kH + h) * KPAD * kA;
  const float* Vm = wsv + (size_t)(bb * kH + h) * KPAD * kA;

  __shared__ float sc[KTOT];
  __shared__ float red[64];
  int t = threadIdx.x;
  const float scale = 0.17677669529663687f;

  float lmax = -1e30f;
  for (int k = t; k < KTOT; k += 64) {
    float d = 0.f;
    for (int a = 0; a < kA; ++a) d += q[a] * Km[(size_t)k * kA + a];
    d *= scale;
    sc[k] = d;
    lmax = fmaxf(lmax, d);
  }
  red[t] = lmax;
  __syncthreads();
  for (int off = 32; off > 0; off >>= 1) {
    if (t < off) red[t] = fmaxf(red[t], red[t + off]);
    __syncthreads();
  }
  float mrow = red[0];
  __syncthreads();

  float lsum = 0.f;
  for (int k = t; k < KTOT; k += 64) {
    float pv = __expf(sc[k] - mrow);
    sc[k] = pv;
    lsum += pv;
  }
  red[t] = lsum;
  __syncthreads();
  for (int off = 32; off > 0; off >>= 1) {
    if (t < off) red[t] += red[t + off];
    __syncthreads();
  }
  float inv = 1.f / red[0];
  __syncthreads();

  if (t < kA) {
    float acc = 0.f;
    for (int k = 0; k < KTOT; ++k) acc += sc[k] * Vm[(size_t)k * kA + t];
    wsoc[((size_t)(bb * kC + c) * kH + h) * kA + t] = acc * inv;
  }
}

// ---------------------------------------------------------------------------
// Kernel 6: head mix (constants): con_out[b,c] (1x64) = out_c (1x256) @ HWc[c]
// grid.x = B*C, 64 threads.
// ---------------------------------------------------------------------------
__global__ void headmix_con_kernel(const float* __restrict__ wsoc,
                                   const float* __restrict__ HWc,
                                   float* __restrict__ out_con) {
  int id = blockIdx.x;
  int bb = id / kC, c = id % kC;
  int e = threadIdx.x;
  const float* oc = wsoc + (size_t)(bb * kC + c) * kF;
  const float* W  = HWc + (size_t)c * kF * kE;
  float acc = 0.f;
  for (int f = 0; f < kF; ++f) acc += oc[f] * W[(size_t)f * kE + e];
  out_con[(size_t)(bb * kC + c) * kE + e] = acc;
}

// ---------------------------------------------------------------------------
extern "C" void kernel_launch(void* const* d_in, const int* in_sizes, int n_in,
                              void* d_out, int out_size, void* d_ws, size_t ws_size,
                              hipStream_t stream) {
  const float* qs  = (const float*)d_in[0];
  const float* ks  = (const float*)d_in[1];
  const float* vs  = (const float*)d_in[2];
  const float* qc  = (const float*)d_in[3];
  const float* kc  = (const float*)d_in[4];
  const float* vc  = (const float*)d_in[5];
  const float* qws = (const float*)d_in[6];
  const float* kws = (const float*)d_in[7];
  const float* vws = (const float*)d_in[8];
  const float* qwc = (const float*)d_in[9];
  const float* kwc = (const float*)d_in[10];
  const float* vwc = (const float*)d_in[11];
  const float* hws = (const float*)d_in[12];
  const float* hwc = (const float*)d_in[13];
  const int*   msk = (const int*)d_in[14];

  float* out = (float*)d_out;
  float* ws  = (float*)d_ws;

  float* wsq  = ws;
  float* wsk  = wsq + WSQ_N;
  float* wsv  = wsk + WSK_N;
  float* wso  = wsv + WSV_N;
  float* wsqc = wso + WSO_N;
  float* wsoc = wsqc + WSQC_N;

  proj_seq_kernel<<<3 * kB * kS * kH, 256, 0, stream>>>(qs, ks, vs, qws, kws, vws,
                                                        wsq, wsk, wsv);
  proj_con_kernel<<<kB * kC * kH, 96, 0, stream>>>(qc, kc, vc, qwc, kwc, vwc,
                                                   wsqc, wsk, wsv);
  attn_seq_kernel<<<kB * kS * kH, 256, 0, stream>>>(wsq, wsk, wsv, wso, msk);
  headmix_seq_kernel<<<kB * kS, 256, 0, stream>>>(wso, hws, out);
  attn_con_kernel<<<kB * kC * kH, 64, 0, stream>>>(wsqc, wsk, wsv, wsoc);
  headmix_con_kernel<<<kB * kC, 64, 0, stream>>>(wsoc, hwc,
                                                 out + (size_t)kB * kS * kL * kE);
}